// DiffAttention_46402826666664
// MI455X (gfx1250) — compile-verified
//
#include <hip/hip_runtime.h>
#include <hip/hip_bf16.h>

// ---------------------------------------------------------------------------
// DiffAttention (GAT-style) for MI455X / gfx1250, wave32.
//   h      = h_init @ W1^T                      (WMMA f32 16x16x4)
//   e      = tanh((h[dst]-h[src]) . a)          (wave-per-edge, shfl reduce)
//   alpha  = segment_softmax(e, dst)            (L2 atomics: max-key + add)
//   out    = relu(h + segsum(alpha*(h[dst]-h[src])))  (atomic f32 scatter)
// ---------------------------------------------------------------------------

#define IN_DIM  128
#define OUT_DIM 64

typedef __attribute__((ext_vector_type(2))) float v2f;
typedef __attribute__((ext_vector_type(8))) float v8f;

// Monotonic unsigned key for float atomicMax (total order incl. negatives).
__device__ __forceinline__ unsigned fkey(float f) {
    unsigned b = __float_as_uint(f);
    return (b & 0x80000000u) ? ~b : (b | 0x80000000u);
}
__device__ __forceinline__ float fkey_inv(unsigned k) {
    unsigned b = (k & 0x80000000u) ? (k ^ 0x80000000u) : ~k;
    return __uint_as_float(b);
}

// ---------------------------------------------------------------------------
// K0: zero-init scratch (h_diff, denom, node max-keys).
// ---------------------------------------------------------------------------
__global__ void k_init(float* __restrict__ hdiff, float* __restrict__ denom,
                       unsigned* __restrict__ nkey, int N) {
    int i = blockIdx.x * blockDim.x + threadIdx.x;
    if (i < N * OUT_DIM) hdiff[i] = 0.0f;
    if (i < N) { denom[i] = 0.0f; nkey[i] = 0u; }  // key(-inf) > 0, so 0 is identity
}

// ---------------------------------------------------------------------------
// K1: h = h_init @ W1^T via V_WMMA_F32_16X16X4_F32.
// Block = 128 threads = 4 waves; wave w owns output-column tile w (16 cols).
// Grid.x = N/16 row tiles. A: lane holds row lane%16, K pair 2*(lane/16).
// B (= W1 tile, since B[k][n] = W1[n_global][k]) mirrors that layout.
// ---------------------------------------------------------------------------
__global__ void k_gemm_wmma(const float* __restrict__ X,   // [N, 128]
                            const float* __restrict__ W,   // [64, 128]
                            float* __restrict__ H,         // [N, 64]
                            int N) {
    const int lane   = threadIdx.x & 31;
    const int wave   = threadIdx.x >> 5;          // 0..3 -> output col tile
    const int m      = lane & 15;                 // A row / B col within tile
    const int khalf  = (lane >> 4) << 1;          // 0 or 2

    const int row0   = blockIdx.x * 16;
    const float* __restrict__ xrow = X + (size_t)(row0 + m) * IN_DIM;
    const float* __restrict__ wrow = W + (size_t)(wave * 16 + m) * IN_DIM;

    v8f acc = {};
#pragma unroll
    for (int k = 0; k < IN_DIM; k += 4) {
        v2f A, B;
        const float2 xa = *(const float2*)(xrow + k + khalf);
        const float2 wb = *(const float2*)(wrow + k + khalf);
        A.x = xa.x; A.y = xa.y;
        B.x = wb.x; B.y = wb.y;
        // 8 args: (neg_a, A, neg_b, B, c_mod, C, reuse_a, reuse_b)
        acc = __builtin_amdgcn_wmma_f32_16x16x4_f32(
            false, A, false, B, (short)0, acc, false, false);
    }

    // D layout: VGPR v -> M = v + 8*(lane>>4), N = lane&15.
    const int col = wave * 16 + (lane & 15);
    const int mrb = row0 + ((lane >> 4) << 3);
#pragma unroll
    for (int v = 0; v < 8; ++v)
        H[(size_t)(mrb + v) * OUT_DIM + col] = acc[v];
}

// ---------------------------------------------------------------------------
// K2: wave per edge: e = tanh((h[dst]-h[src]) . a); atomicMax key into dst.
// Each lane covers 2 of the 64 dims -> one coalesced 256B read per row.
// ---------------------------------------------------------------------------
__global__ void k_edge_logit(const float* __restrict__ H,
                             const float* __restrict__ a,
                             const int* __restrict__ src,
                             const int* __restrict__ dst,
                             float* __restrict__ e,
                             unsigned* __restrict__ nkey, int E) {
    const int edge = blockIdx.x * (blockDim.x >> 5) + (threadIdx.x >> 5);
    if (edge >= E) return;
    const int lane = threadIdx.x & 31;
    const int s = src[edge], d = dst[edge];

    const float2 hd = *(const float2*)(H + (size_t)d * OUT_DIM + lane * 2);
    const float2 hs = *(const float2*)(H + (size_t)s * OUT_DIM + lane * 2);
    const float2 av = *(const float2*)(a + lane * 2);
    float part = (hd.x - hs.x) * av.x + (hd.y - hs.y) * av.y;

#pragma unroll
    for (int off = 16; off > 0; off >>= 1)
        part += __shfl_down(part, off, 32);

    if (lane == 0) {
        const float ev = tanhf(part);
        e[edge] = ev;
        atomicMax(nkey + d, fkey(ev));
    }
}

// ---------------------------------------------------------------------------
// K3: thread per edge: ex = exp(e - max[dst]); denom[dst] += ex (in place e->ex).
// ---------------------------------------------------------------------------
__global__ void k_edge_exp(const int* __restrict__ dst,
                           float* __restrict__ e,
                           const unsigned* __restrict__ nkey,
                           float* __restrict__ denom, int E) {
    const int i = blockIdx.x * blockDim.x + threadIdx.x;
    if (i >= E) return;
    const int d = dst[i];
    const float ex = __expf(e[i] - fkey_inv(nkey[d]));
    e[i] = ex;
    atomicAdd(denom + d, ex);
}

// ---------------------------------------------------------------------------
// K4: wave per edge: h_diff[dst] += (ex/denom[dst]) * (h[dst]-h[src]).
// ---------------------------------------------------------------------------
__global__ void k_edge_scatter(const float* __restrict__ H,
                               const int* __restrict__ src,
                               const int* __restrict__ dst,
                               const float* __restrict__ e,
                               const float* __restrict__ denom,
                               float* __restrict__ hdiff, int E) {
    const int edge = blockIdx.x * (blockDim.x >> 5) + (threadIdx.x >> 5);
    if (edge >= E) return;
    const int lane = threadIdx.x & 31;
    const int s = src[edge], d = dst[edge];
    const float alpha = e[edge] / denom[d];

    const float2 hd = *(const float2*)(H + (size_t)d * OUT_DIM + lane * 2);
    const float2 hs = *(const float2*)(H + (size_t)s * OUT_DIM + lane * 2);
    float* p = hdiff + (size_t)d * OUT_DIM + lane * 2;
    atomicAdd(p + 0, alpha * (hd.x - hs.x));
    atomicAdd(p + 1, alpha * (hd.y - hs.y));
}

// ---------------------------------------------------------------------------
// K5: out = relu(h + h_diff)   (h already lives in d_out)
// ---------------------------------------------------------------------------
__global__ void k_finish(float* __restrict__ out, const float* __restrict__ hdiff,
                         int total) {
    const int i = blockIdx.x * blockDim.x + threadIdx.x;
    if (i >= total) return;
    const float v = out[i] + hdiff[i];
    out[i] = v > 0.0f ? v : 0.0f;
}

// ---------------------------------------------------------------------------
extern "C" void kernel_launch(void* const* d_in, const int* in_sizes, int n_in,
                              void* d_out, int out_size, void* d_ws, size_t ws_size,
                              hipStream_t stream) {
    const float* h_init = (const float*)d_in[0];   // [N,128]
    const float* W1     = (const float*)d_in[1];   // [64,128]
    const float* a      = (const float*)d_in[2];   // [1,64]
    const int*   src    = (const int*)d_in[3];     // [E]
    const int*   dst    = (const int*)d_in[4];     // [E]

    const int N = in_sizes[0] / IN_DIM;
    const int E = in_sizes[3];

    // Workspace layout (floats/uints, 4B aligned):
    //   hdiff[N*64] | e[E] | denom[N] | nkey[N]
    float*    hdiff = (float*)d_ws;
    float*    e     = hdiff + (size_t)N * OUT_DIM;
    float*    denom = e + E;
    unsigned* nkey  = (unsigned*)(denom + N);

    float* H = (float*)d_out;                      // h lives in d_out

    // K0: init scratch
    {
        const int total = N * OUT_DIM;
        k_init<<<(total + 255) / 256, 256, 0, stream>>>(hdiff, denom, nkey, N);
    }
    // K1: WMMA GEMM (N assumed multiple of 16: 100000 = 6250*16)
    k_gemm_wmma<<<N / 16, 128, 0, stream>>>(h_init, W1, H, N);

    // K2: edge logits + segment max (8 edges per 256-thread block)
    k_edge_logit<<<(E + 7) / 8, 256, 0, stream>>>(H, a, src, dst, e, nkey, E);

    // K3: exp + segment sum
    k_edge_exp<<<(E + 255) / 256, 256, 0, stream>>>(dst, e, nkey, denom, E);

    // K4: weighted scatter
    k_edge_scatter<<<(E + 7) / 8, 256, 0, stream>>>(H, src, dst, e, denom, hdiff, E);

    // K5: fused add + relu
    {
        const int total = N * OUT_DIM;
        k_finish<<<(total + 255) / 256, 256, 0, stream>>>(H, hdiff, total);
    }
}